// pendelum2dof_49203145343518
// MI455X (gfx1250) — compile-verified
//
#include <hip/hip_runtime.h>
#include <math.h>

// Double-pendulum Hamiltonian dynamics, closed form.
// Constants from the reference: M1 = M2 = L1 = L2 = 1, G = 9.81.
// State per row: (th1, th2, p1, p2)  ->  output (dth1, dth2, dp1, dp2).
//
// Derivation (m1=m2=l1=l2=1):
//   Delta = th1 - th2, c = cos(Delta), s = sin(Delta)
//   Mass matrix M = [[2, c], [c, 1]]  (Hessian of kinetic energy; momentum-independent)
//   D = det(M) = 2 - c^2
//   dth = M^{-1} p:
//     dth1 = (p1 - c*p2) / D
//     dth2 = (2*p2 - c*p1) / D
//   T(theta, p) = (p1^2 - 2 c p1 p2 + 2 p2^2) / (2 D) = N / (2D)
//   dT/dDelta = s * (p1*p2*D - c*N) / D^2          (= dT/dth1 = -dT/dth2)
//   V' : dV/dth1 = 2 G sin(th1),  dV/dth2 = G sin(th2)
//   dp1 = -dT/dth1 - 2 G sin(th1)
//   dp2 = +dT/dth1 -   G sin(th2)

#define DP_G 9.81f

__global__ __launch_bounds__(256) void dp2dof_dyn_kernel(
    const float4* __restrict__ y, float4* __restrict__ out, int n) {
  int i = blockIdx.x * blockDim.x + threadIdx.x;
  if (i >= n) return;

  // One state per lane: coalesced global_load_b128 (512B per wave32).
  float4 v = y[i];
  float th1 = v.x, th2 = v.y, p1 = v.z, p2 = v.w;

  float s, c;
  sincosf(th1 - th2, &s, &c);

  float D    = fmaf(-c, c, 2.0f);        // 2 - c^2  (>= 1, division is safe)
  float invD = 1.0f / D;

  // dq = M^{-1} p
  float dth1 = fmaf(-c, p2, p1) * invD;          // (p1 - c*p2)/D
  float dth2 = fmaf(-c, p1, 2.0f * p2) * invD;   // (2*p2 - c*p1)/D

  // Kinetic torque term: s*(p1*p2*D - c*N)/D^2
  float p12 = p1 * p2;
  float N   = fmaf(p1, p1, fmaf(2.0f * p2, p2, -2.0f * c * p12));
  float tq  = s * fmaf(p12, D, -c * N) * invD * invD;

  float dp1 = -tq - 2.0f * DP_G * sinf(th1);
  float dp2 =  tq -        DP_G * sinf(th2);

  out[i] = make_float4(dth1, dth2, dp1, dp2);
}

extern "C" void kernel_launch(void* const* d_in, const int* in_sizes, int n_in,
                              void* d_out, int out_size, void* d_ws, size_t ws_size,
                              hipStream_t stream) {
  // d_in[0] = t (1 element, unused ODE-time argument)
  // d_in[1] = y (B*4 float32, row-major (B,4))
  const float4* y = (const float4*)d_in[1];
  float4* out = (float4*)d_out;

  int n = in_sizes[1] / 4;  // number of states (B)
  const int block = 256;    // 8 waves per block on wave32
  int grid = (n + block - 1) / block;

  dp2dof_dyn_kernel<<<grid, block, 0, stream>>>(y, out, n);
}